// SelfAttention_with_gos_and_masking_40716289966634
// MI455X (gfx1250) — compile-verified
//
#include <hip/hip_runtime.h>
#include <hip/hip_bf16.h>

// ---------------------------------------------------------------------------
// Self-attention w/ masking, fp32, for gfx1250 using V_WMMA_F32_16X16X4_F32
// plus GLOBAL_LOAD_ASYNC_TO_LDS_B128 for tile staging.
// B=4, N=2048, D=512, H=8, DH=64.
// ---------------------------------------------------------------------------

typedef __attribute__((ext_vector_type(2))) float v2f;
typedef __attribute__((ext_vector_type(8))) float v8f;

#define WMMA_F32(a, b, c) \
  __builtin_amdgcn_wmma_f32_16x16x4_f32(false, (a), false, (b), (short)0, (c), false, false)

// Async global->LDS b128 copy (per enabled lane), tracked by ASYNCcnt.
__device__ __forceinline__ void async_copy_b128(const float* gsrc, float* ldst) {
  unsigned lds_off = (unsigned)(uintptr_t)ldst;
  unsigned long long gaddr = (unsigned long long)(uintptr_t)gsrc;
  asm volatile("global_load_async_to_lds_b128 %0, %1, off"
               :: "v"(lds_off), "v"(gaddr) : "memory");
}
__device__ __forceinline__ void async_wait0() {
  asm volatile("s_wait_asynccnt 0x0" ::: "memory");
}

#define B_ 4
#define N_ 2048
#define D_ 512
#define H_ 8
#define DH_ 64
#define NEG_BIG_ (-1.0e10f)

// ---------------------------------------------------------------------------
// Generic tiled GEMM: C[M,Nn] = A[M,K] * B[K,Nn] (+ bias[Nn]).
// Block tile 64x64, 4 waves each owning a 32x32 quadrant (2x2 WMMA tiles).
// fp32 WMMA 16x16x4; K staged 16 wide via async b128 into padded LDS.
// Strides: As row = 20 floats (80B, 16B-aligned; 16-lane column reads hit 16
// distinct banks, gcd(20,64)=4), Bs row = 68 floats (272B = 17*16B).
// ---------------------------------------------------------------------------
__global__ __launch_bounds__(128) void gemm_wmma_f32(
    const float* __restrict__ A, const float* __restrict__ Bm,
    const float* __restrict__ bias, float* __restrict__ C,
    int M, int Nn, int K) {
  __shared__ __align__(16) float As[64][20];   // [row][k]
  __shared__ __align__(16) float Bs[16][68];   // [k][col]
  const int tid  = threadIdx.x;
  const int lane = tid & 31;
  const int wave = tid >> 5;
  const int wr = (wave >> 1) * 32;
  const int wc = (wave & 1) * 32;
  const int blockRow = blockIdx.y * 64;
  const int blockCol = blockIdx.x * 64;
  const int lh = lane >> 4;   // half-wave select -> K pair (A/B) / row group (C)
  const int ll = lane & 15;

  v8f acc[2][2] = {};

  for (int kt = 0; kt < K; kt += 16) {
    __syncthreads();
    // A tile 64x16 + B tile 16x64: 256 float4 each, 2 per thread (fixed trip)
#pragma unroll
    for (int t = 0; t < 2; ++t) {
      int i = tid + t * 128;
      int r = i >> 2, k4 = (i & 3) * 4;
      async_copy_b128(&A[(size_t)(blockRow + r) * K + kt + k4], &As[r][k4]);
      int kk = i >> 4, c4 = (i & 15) * 4;
      async_copy_b128(&Bm[(size_t)(kt + kk) * Nn + blockCol + c4], &Bs[kk][c4]);
    }
    async_wait0();
    __syncthreads();
#pragma unroll
    for (int k4 = 0; k4 < 4; ++k4) {
      const int ko = k4 * 4 + lh * 2;
      v2f a0, a1, b0, b1;
      a0.x = As[wr + ll][ko];       a0.y = As[wr + ll][ko + 1];
      a1.x = As[wr + 16 + ll][ko];  a1.y = As[wr + 16 + ll][ko + 1];
      b0.x = Bs[ko][wc + ll];       b0.y = Bs[ko + 1][wc + ll];
      b1.x = Bs[ko][wc + 16 + ll];  b1.y = Bs[ko + 1][wc + 16 + ll];
      acc[0][0] = WMMA_F32(a0, b0, acc[0][0]);
      acc[0][1] = WMMA_F32(a0, b1, acc[0][1]);
      acc[1][0] = WMMA_F32(a1, b0, acc[1][0]);
      acc[1][1] = WMMA_F32(a1, b1, acc[1][1]);
    }
  }
  // epilogue: bias hoisted per output column (uniform over the 8 acc rows)
#pragma unroll
  for (int ni = 0; ni < 2; ++ni) {
    const int col = blockCol + wc + ni * 16 + ll;
    const float bv = bias ? bias[col] : 0.0f;
#pragma unroll
    for (int mi = 0; mi < 2; ++mi)
#pragma unroll
      for (int v = 0; v < 8; ++v) {
        int row = blockRow + wr + mi * 16 + v + 8 * lh;
        C[(size_t)row * Nn + col] = acc[mi][ni][v] + bv;
      }
  }
}

// ---------------------------------------------------------------------------
// Flash-attention: one block = (b, h, 64 query rows); 4 waves x 16 rows each.
// K/V tiles (16x64) staged async into LDS; scores + O-update via fp32 WMMA;
// online softmax with 16-lane shuffle reductions (rows live in 16-lane
// halves per the WMMA C layout). ctx layout: [B, N, H*DH].
// ---------------------------------------------------------------------------
__global__ __launch_bounds__(128) void attn_flash(
    const float* __restrict__ qkv, const float* __restrict__ mask,
    float* __restrict__ ctx) {
  __shared__ __align__(16) float Kt[16][68];     // [col c][d]
  __shared__ __align__(16) float Vt[16][68];     // [col c][d]
  __shared__ __align__(16) float Ps[4][16][20];  // per-wave P: C->A layout
  const int tid = threadIdx.x, lane = tid & 31, wave = tid >> 5;
  const int lh = lane >> 4, ll = lane & 15;
  const int b = blockIdx.z, h = blockIdx.y;
  const int rowBase = blockIdx.x * 64 + wave * 16;
  const float scale = 0.125f;  // DH^-0.5

  // Q A-fragments: 16 k-steps of 4 over DH=64
  v2f qa[16];
  {
    const float* qrow = qkv + (size_t)(b * N_ + rowBase + ll) * (3 * H_ * DH_) + h * DH_;
#pragma unroll
    for (int ks = 0; ks < 16; ++ks) {
      int d = ks * 4 + lh * 2;
      qa[ks].x = qrow[d];
      qa[ks].y = qrow[d + 1];
    }
  }

  float mrow[8], lrow[8];
#pragma unroll
  for (int v = 0; v < 8; ++v) { mrow[v] = -3.0e38f; lrow[v] = 0.0f; }
  v8f o[4] = {};

  for (int j = 0; j < N_ / 16; ++j) {
    __syncthreads();
    // K/V tiles: 16x64 each = 256 float4 each; 2 float4 per thread per tile
#pragma unroll
    for (int t = 0; t < 2; ++t) {
      int i = tid + t * 128;
      int c = i >> 4, d4 = (i & 15) * 4;
      size_t base = (size_t)(b * N_ + j * 16 + c) * (3 * H_ * DH_) + h * DH_ + d4;
      async_copy_b128(&qkv[base + H_ * DH_],     &Kt[c][d4]);
      async_copy_b128(&qkv[base + 2 * H_ * DH_], &Vt[c][d4]);
    }
    async_wait0();
    __syncthreads();

    // S = Q * K^T  (B[d][c] = Kt[c][d])
    v8f s = {};
#pragma unroll
    for (int ks = 0; ks < 16; ++ks) {
      int ko = ks * 4 + lh * 2;
      v2f bf;
      bf.x = Kt[ll][ko];
      bf.y = Kt[ll][ko + 1];
      s = WMMA_F32(qa[ks], bf, s);
    }

    // scale, mask, online softmax
    const int col = j * 16 + ll;
#pragma unroll
    for (int v = 0; v < 8; ++v) {
      int row = rowBase + v + 8 * lh;
      float sv = s[v] * scale;
      float mk = mask[(size_t)(b * N_ + row) * N_ + col];
      sv = (mk == 1.0f) ? NEG_BIG_ : sv;
      float rm = sv;
      rm = fmaxf(rm, __shfl_xor(rm, 1, 32));
      rm = fmaxf(rm, __shfl_xor(rm, 2, 32));
      rm = fmaxf(rm, __shfl_xor(rm, 4, 32));
      rm = fmaxf(rm, __shfl_xor(rm, 8, 32));
      float mnew  = fmaxf(mrow[v], rm);
      float alpha = __expf(mrow[v] - mnew);
      float p     = __expf(sv - mnew);
      float rs = p;
      rs += __shfl_xor(rs, 1, 32);
      rs += __shfl_xor(rs, 2, 32);
      rs += __shfl_xor(rs, 4, 32);
      rs += __shfl_xor(rs, 8, 32);
      lrow[v] = lrow[v] * alpha + rs;
      mrow[v] = mnew;
      s[v] = p;
#pragma unroll
      for (int t = 0; t < 4; ++t) o[t][v] *= alpha;
    }

    // P: C-layout -> A-layout through per-wave LDS, then O += P * V
#pragma unroll
    for (int v = 0; v < 8; ++v) Ps[wave][v + 8 * lh][ll] = s[v];
    __syncthreads();
#pragma unroll
    for (int k2 = 0; k2 < 4; ++k2) {
      int ko = k2 * 4 + lh * 2;
      v2f af;
      af.x = Ps[wave][ll][ko];
      af.y = Ps[wave][ll][ko + 1];
#pragma unroll
      for (int t = 0; t < 4; ++t) {
        v2f bf;
        bf.x = Vt[ko][t * 16 + ll];
        bf.y = Vt[ko + 1][t * 16 + ll];
        o[t] = WMMA_F32(af, bf, o[t]);
      }
    }
  }

  // normalize and store context
#pragma unroll
  for (int v = 0; v < 8; ++v) {
    int row = rowBase + v + 8 * lh;
    float inv = 1.0f / lrow[v];
    float* crow = ctx + (size_t)(b * N_ + row) * (H_ * DH_) + h * DH_;
#pragma unroll
    for (int t = 0; t < 4; ++t) crow[t * 16 + ll] = o[t][v] * inv;
  }
}

// ---------------------------------------------------------------------------
// Row-0 head-averaged attention + iterative selection of the 17 smallest
// (rank 0 is forced to index 0 via -inf). what_to_prune[j] = rank (16-j).
// One block per batch, 256 threads.
// ---------------------------------------------------------------------------
__global__ __launch_bounds__(256) void row0_prune(
    const float* __restrict__ qkv, const float* __restrict__ mask,
    float* __restrict__ att0, int* __restrict__ prune_out) {
  __shared__ float q0[64];
  __shared__ float red[256];
  __shared__ int   redi[256];
  __shared__ float attL[N_];
  __shared__ int   ranks[17];
  const int b = blockIdx.x;
  const int tid = threadIdx.x;

  for (int i = tid; i < N_; i += 256) attL[i] = 0.0f;

  float vals[8];
  for (int h = 0; h < H_; ++h) {
    __syncthreads();
    if (tid < 64) q0[tid] = qkv[(size_t)(b * N_) * (3 * H_ * DH_) + h * DH_ + tid];
    __syncthreads();
    float lmax = -3.0e38f;
    for (int i = 0; i < 8; ++i) {
      int m = tid + i * 256;
      const float* krow = qkv + (size_t)(b * N_ + m) * (3 * H_ * DH_) + H_ * DH_ + h * DH_;
      float dot = 0.0f;
      for (int d = 0; d < DH_; ++d) dot += q0[d] * krow[d];
      dot *= 0.125f;
      float mk = mask[(size_t)(b * N_) * N_ + m];  // row 0 of mask
      dot = (mk == 1.0f) ? NEG_BIG_ : dot;
      vals[i] = dot;
      lmax = fmaxf(lmax, dot);
    }
    red[tid] = lmax; __syncthreads();
    for (int s = 128; s > 0; s >>= 1) {
      if (tid < s) red[tid] = fmaxf(red[tid], red[tid + s]);
      __syncthreads();
    }
    float gmax = red[0]; __syncthreads();
    float lsum = 0.0f;
    for (int i = 0; i < 8; ++i) { vals[i] = __expf(vals[i] - gmax); lsum += vals[i]; }
    red[tid] = lsum; __syncthreads();
    for (int s = 128; s > 0; s >>= 1) {
      if (tid < s) red[tid] += red[tid + s];
      __syncthreads();
    }
    float gsum = red[0]; __syncthreads();
    for (int i = 0; i < 8; ++i) attL[tid + i * 256] += vals[i] / (gsum * (float)H_);
  }
  __syncthreads();
  if (tid == 0) attL[0] = -3.0e38f;  // reference: att_topk[:,0] = -inf
  __syncthreads();
  for (int i = tid; i < N_; i += 256) att0[b * N_ + i] = attL[i];
  __syncthreads();

  for (int r = 0; r < 17; ++r) {
    float bv = 3.0e38f; int bi = 0x7fffffff;
    for (int i = 0; i < 8; ++i) {
      int m = tid + i * 256;
      float vv = attL[m];
      if (vv < bv || (vv == bv && m < bi)) { bv = vv; bi = m; }
    }
    red[tid] = bv; redi[tid] = bi; __syncthreads();
    for (int s = 128; s > 0; s >>= 1) {
      if (tid < s) {
        float ov = red[tid + s]; int oi = redi[tid + s];
        if (ov < red[tid] || (ov == red[tid] && oi < redi[tid])) {
          red[tid] = ov; redi[tid] = oi;
        }
      }
      __syncthreads();
    }
    if (tid == 0) { ranks[r] = redi[0]; attL[redi[0]] = 3.0e38f; }
    __syncthreads();
  }
  if (tid < 16) prune_out[b * 16 + tid] = ranks[16 - tid];
}

// what_to_merge(4) + how_to_merge(4) = -1; survived_mask(64) = 0
__global__ void fill_tail(int* __restrict__ out_i) {
  int tid = threadIdx.x;
  if (tid < 8) out_i[tid] = -1;
  else if (tid < 72) out_i[tid] = 0;
}

extern "C" void kernel_launch(void* const* d_in, const int* in_sizes, int n_in,
                              void* d_out, int out_size, void* d_ws, size_t ws_size,
                              hipStream_t stream) {
  (void)in_sizes; (void)n_in; (void)out_size; (void)ws_size;
  const float* x    = (const float*)d_in[0];
  const float* mask = (const float*)d_in[1];
  const float* Wqkv = (const float*)d_in[2];
  const float* Wout = (const float*)d_in[3];
  const float* bout = (const float*)d_in[4];
  float* out = (float*)d_out;

  float* qkv  = (float*)d_ws;                               // B*N*1536
  float* ctx  = qkv + (size_t)B_ * N_ * 3 * H_ * DH_;       // B*N*512
  float* att0 = ctx + (size_t)B_ * N_ * H_ * DH_;           // B*N

  // 1) qkv = x @ W_qkv            (8192 x 1536, K=512)
  gemm_wmma_f32<<<dim3((3 * H_ * DH_) / 64, (B_ * N_) / 64), 128, 0, stream>>>(
      x, Wqkv, nullptr, qkv, B_ * N_, 3 * H_ * DH_, D_);

  // 2) masked flash attention -> ctx (B, N, H*DH)
  attn_flash<<<dim3(N_ / 64, H_, B_), 128, 0, stream>>>(qkv, mask, ctx);

  // 3) row-0 head-mean attention + prune index selection
  const size_t OUT_F = (size_t)B_ * N_ * D_;   // 4194304 floats
  int* prune = (int*)d_out + OUT_F;
  row0_prune<<<B_, 256, 0, stream>>>(qkv, mask, att0, prune);

  // 4) out = ctx @ W_out + b_out  (8192 x 512, K=512)
  gemm_wmma_f32<<<dim3(D_ / 64, (B_ * N_) / 64), 128, 0, stream>>>(
      ctx, Wout, bout, out, B_ * N_, D_, H_ * DH_);

  // 5) constant tail outputs
  fill_tail<<<1, 128, 0, stream>>>((int*)d_out + OUT_F + B_ * 16);
}